// SelectiveSearch_71768903516381
// MI455X (gfx1250) — compile-verified
//
#include <hip/hip_runtime.h>

// ---------------- problem constants (from reference setup_inputs) ----------
#define S_   1024
#define CB_  25
#define TB_  10
#define B_   8
#define C_   3
#define R_   8
#define H_   512
#define W_   512
#define HW_  (H_ * W_)
#define EPS_ 1e-12f

// ---------------- d_out layout (floats, reference return order) ------------
#define OFF_AREA  0
#define N_AREA    (B_ * S_)                       // 8192
#define OFF_BBOX  (OFF_AREA + N_AREA)             // 8192
#define N_BBOX    (B_ * S_ * 4)                   // 32768
#define OFF_COLOR (OFF_BBOX + N_BBOX)             // 40960
#define N_COLOR   (B_ * S_ * C_ * CB_)            // 614400
#define OFF_TEX   (OFF_COLOR + N_COLOR)           // 655360
#define N_TEX     (B_ * S_ * C_ * R_ * TB_)       // 1966080
#define OFF_GRAD  (OFF_TEX + N_TEX)               // 2621440

// ws (int32): area | xmin | ymin | xmax | ymax, each B*S
#define WS_AREA 0
#define WS_XMIN (1 * B_ * S_)
#define WS_YMIN (2 * B_ * S_)
#define WS_XMAX (3 * B_ * S_)
#define WS_YMAX (4 * B_ * S_)
#define WS_INTS (5 * B_ * S_)

typedef unsigned int u32x4 __attribute__((ext_vector_type(4)));
typedef unsigned int u32x8 __attribute__((ext_vector_type(8)));

// ------------- CDNA5 async global->LDS (inline asm, ASYNCcnt path) --------
__device__ __forceinline__ void async_g2l_b32(const float* gptr, const float* lptr) {
  // Low 32 bits of a flat shared-aperture address are the LDS byte address
  // (ISA 10.2).
  unsigned lds_addr = (unsigned)(unsigned long long)lptr;
  unsigned long long gaddr = (unsigned long long)gptr;
  asm volatile("global_load_async_to_lds_b32 %0, %1, off"
               :: "v"(lds_addr), "v"(gaddr)
               : "memory");
}

__device__ __forceinline__ void async_wait0() {
  asm volatile("s_wait_asynccnt 0" ::: "memory");
}

__device__ __forceinline__ void tensor_wait0() {
#if __has_builtin(__builtin_amdgcn_s_wait_tensorcnt)
  __builtin_amdgcn_s_wait_tensorcnt(0);
#else
  asm volatile("s_wait_tensorcnt 0x0" ::: "memory");
#endif
}

// ---------------- 1) init: zero hists, seed ws sentinels -------------------
__global__ void k_init(float* __restrict__ out, int* __restrict__ ws) {
  const int tid = blockIdx.x * blockDim.x + threadIdx.x;
  const int stride = gridDim.x * blockDim.x;
  float4* h4 = (float4*)(out + OFF_COLOR);            // offset 40960: 16B aligned
  const int n4 = (N_COLOR + N_TEX) / 4;
  const float4 z4 = make_float4(0.f, 0.f, 0.f, 0.f);
  for (int i = tid; i < n4; i += stride) h4[i] = z4;
  for (int i = tid; i < WS_INTS; i += stride)
    ws[i] = (i >= WS_XMIN && i < WS_XMAX) ? W_ : 0;   // xmin=ymin=512, rest 0
}

// ---------------- 2) area + bbox (LDS privatized, b128 label loads) --------
#define ACH 16  // chunks per image
__global__ void k_area_bbox(const int* __restrict__ lab, int* __restrict__ ws) {
  __shared__ int sA[S_], sXn[S_], sYn[S_], sXx[S_], sYx[S_];
  const int b = blockIdx.x / ACH;
  const int chunk = blockIdx.x % ACH;
  for (int i = threadIdx.x; i < S_; i += blockDim.x) {
    sA[i] = 0; sXn[i] = W_; sYn[i] = H_; sXx[i] = 0; sYx[i] = 0;
  }
  __syncthreads();
  const int npx4 = (HW_ / ACH) / 4;
  const int q0 = chunk * npx4;
  const int4* lab4 = (const int4*)(lab + b * HW_);
  for (int q = threadIdx.x; q < npx4; q += blockDim.x) {
    const int i4 = q0 + q;
    const int4 L = lab4[i4];
    const int ls[4] = {L.x, L.y, L.z, L.w};
#pragma unroll
    for (int j = 0; j < 4; ++j) {
      const int px = i4 * 4 + j;
      const int l = ls[j];
      const int x = px & (W_ - 1);
      const int y = px >> 9;  // W_ == 512
      atomicAdd(&sA[l], 1);
      atomicMin(&sXn[l], x);
      atomicMin(&sYn[l], y);
      atomicMax(&sXx[l], x);
      atomicMax(&sYx[l], y);
    }
  }
  __syncthreads();
  for (int i = threadIdx.x; i < S_; i += blockDim.x) {
    if (sA[i] > 0) {
      atomicAdd(&ws[WS_AREA + b * S_ + i], sA[i]);
      atomicMin(&ws[WS_XMIN + b * S_ + i], sXn[i]);
      atomicMin(&ws[WS_YMIN + b * S_ + i], sYn[i]);
      atomicMax(&ws[WS_XMAX + b * S_ + i], sXx[i]);
      atomicMax(&ws[WS_YMAX + b * S_ + i], sYx[i]);
    }
  }
}

// ---------------- 3) color histogram (100KB LDS per block) -----------------
#define CCH 4  // chunks per (b,c) plane
__global__ void k_color(const int* __restrict__ lab, const int* __restrict__ bins,
                        float* __restrict__ out) {
  extern __shared__ unsigned int shc[];  // S_*CB_ = 25600 u32 = 100KB
  const int chunk = blockIdx.x % CCH;
  const int c = (blockIdx.x / CCH) % C_;
  const int b = blockIdx.x / (CCH * C_);
  for (int i = threadIdx.x; i < S_ * CB_; i += blockDim.x) shc[i] = 0u;
  __syncthreads();
  const int npx4 = (HW_ / CCH) / 4;
  const int q0 = chunk * npx4;
  const int4* lab4 = (const int4*)(lab + b * HW_);
  const int4* bin4 = (const int4*)(bins + (b * C_ + c) * HW_);
  for (int q = threadIdx.x; q < npx4; q += blockDim.x) {
    const int i4 = q0 + q;
    const int4 L = lab4[i4];
    const int4 K = bin4[i4];
    const int ls[4] = {L.x, L.y, L.z, L.w};
    const int ks[4] = {K.x, K.y, K.z, K.w};
#pragma unroll
    for (int j = 0; j < 4; ++j) atomicAdd(&shc[ls[j] * CB_ + ks[j]], 1u);
  }
  __syncthreads();
  float* oc = out + OFF_COLOR;
  for (int i = threadIdx.x; i < S_ * CB_; i += blockDim.x) {
    const unsigned int v = shc[i];
    if (v) {
      const int s = i / CB_, k = i % CB_;
      atomicAdd(&oc[(b * S_ + s) * (C_ * CB_) + c * CB_ + k], (float)v);
    }
  }
}

// ---------------- 4) texture histogram (2 r-planes, 80KB LDS) --------------
#define TCH 2  // pixel chunks per (b,c,r-pair)
__global__ void k_texture(const int* __restrict__ lab, const int* __restrict__ bins,
                          float* __restrict__ out) {
  extern __shared__ unsigned int sht[];  // 2 * S_*TB_ = 20480 u32 = 80KB
  const int chunk = blockIdx.x % TCH;
  const int rg = (blockIdx.x / TCH) % (R_ / 2);
  const int c = (blockIdx.x / (TCH * (R_ / 2))) % C_;
  const int b = blockIdx.x / (TCH * (R_ / 2) * C_);
  const int r0 = rg * 2;
  for (int i = threadIdx.x; i < 2 * S_ * TB_; i += blockDim.x) sht[i] = 0u;
  __syncthreads();
  const int npx4 = (HW_ / TCH) / 4;
  const int q0 = chunk * npx4;
  const int4* lab4 = (const int4*)(lab + b * HW_);
  const int4* bin4 = (const int4*)(bins + ((b * C_ + c) * R_ + r0) * HW_);
  for (int q = threadIdx.x; q < npx4; q += blockDim.x) {
    const int i4 = q0 + q;
    const int4 L = lab4[i4];
    const int4 K0 = bin4[i4];
    const int4 K1 = bin4[i4 + HW_ / 4];
    const int ls[4] = {L.x, L.y, L.z, L.w};
    const int k0s[4] = {K0.x, K0.y, K0.z, K0.w};
    const int k1s[4] = {K1.x, K1.y, K1.z, K1.w};
#pragma unroll
    for (int j = 0; j < 4; ++j) {
      atomicAdd(&sht[ls[j] * TB_ + k0s[j]], 1u);
      atomicAdd(&sht[S_ * TB_ + ls[j] * TB_ + k1s[j]], 1u);
    }
  }
  __syncthreads();
  float* ot = out + OFF_TEX;
  for (int i = threadIdx.x; i < 2 * S_ * TB_; i += blockDim.x) {
    const unsigned int v = sht[i];
    if (v) {
      const int rr = i / (S_ * TB_);
      const int k = i % (S_ * TB_);
      const int s = k / TB_, t = k % TB_;
      atomicAdd(&ot[(b * S_ + s) * (C_ * R_ * TB_) + c * (R_ * TB_) + (r0 + rr) * TB_ + t],
                (float)v);
    }
  }
}

// ---------------- 5) finalize area + bbox(xywh) ----------------------------
__global__ void k_finalize(const int* __restrict__ ws, float* __restrict__ out) {
  const int i = blockIdx.x * blockDim.x + threadIdx.x;
  if (i < B_ * S_) {
    const int xmin = ws[WS_XMIN + i], ymin = ws[WS_YMIN + i];
    const int xmax = ws[WS_XMAX + i], ymax = ws[WS_YMAX + i];
    out[OFF_AREA + i] = (float)ws[WS_AREA + i];
    float* bb = out + OFF_BBOX + i * 4;
    bb[0] = (float)xmin;
    bb[1] = (float)ymin;
    bb[2] = (float)(xmax - xmin);  // reference: no +1
    bb[3] = (float)(ymax - ymin);
  }
}

// ---------------- 6) normalize histograms ----------------------------------
__global__ void k_normalize(const int* __restrict__ ws, float* __restrict__ out) {
  const int tid = blockIdx.x * blockDim.x + threadIdx.x;
  const int stride = gridDim.x * blockDim.x;
  for (int idx = tid; idx < N_COLOR + N_TEX; idx += stride) {
    int b, s;
    float mult;
    if (idx < N_COLOR) {
      b = idx / (S_ * C_ * CB_);
      s = (idx / (C_ * CB_)) % S_;
      mult = (float)C_;
    } else {
      const int j = idx - N_COLOR;
      b = j / (S_ * C_ * R_ * TB_);
      s = (j / (C_ * R_ * TB_)) % S_;
      mult = (float)(C_ * R_);
    }
    const float a = (float)ws[WS_AREA + b * S_ + s];
    out[OFF_COLOR + idx] /= (a * mult + EPS_);
  }
}

// ---------------- 7) Scharr gradients: TDM tiles + async fallback ----------
// Output tile: 62x16.  Input tile in LDS: 18 rows x 64 cols (contiguous, so a
// single 2D TENSOR_LOAD_TO_LDS with tile_dim0=64, tile_dim1=18, stride=512
// lands it with no padding).  Border blocks use per-lane async-LDS + pre-zero.
#define CTW 62
#define CTH 16
#define LTW 64
#define LTH 18
#define NBX 9   // ceil(512/62)
#define NBY 32  // 512/16

__global__ void k_scharr(const float* __restrict__ img, float* __restrict__ out) {
  __shared__ float tile[LTH * LTW];  // 1152 floats = 4.5KB
  const int bpp = NBX * NBY;         // 288 blocks per plane
  const int plane = blockIdx.x / bpp;
  const int rem = blockIdx.x % bpp;
  const int ty = rem / NBX, tx = rem % NBX;
  const int y0 = ty * CTH, x0 = tx * CTW;
  const int gy0 = y0 - 1, gx0 = x0 - 1;
  const float* src = img + plane * HW_;
  const bool interior = (tx >= 1) && (tx <= 7) && (ty >= 1) && (ty <= 30);

  if (interior) {
    // One TDM descriptor per block, issued by wave 0 only (TDM ignores EXEC,
    // so gate with a compiler-visible uniform branch to avoid duplicate DMAs).
    if (__builtin_amdgcn_readfirstlane((int)(threadIdx.x >> 5)) == 0) {
      const unsigned long long ga = (unsigned long long)(src + gy0 * W_ + gx0);
      const unsigned lds = (unsigned)(unsigned long long)(&tile[0]);
      // D# group 0: count=1 | lds_addr | global_addr[56:0] | type=2
      u32x4 g0 = {1u, lds, (unsigned)ga,
                  (unsigned)((ga >> 32) & 0x1FFFFFFu) | (2u << 30)};
      // D# group 1: data_size=4B; tensor_dim0=512, tensor_dim1=big;
      // tile_dim0=64, tile_dim1=18, tile_dim2=0; dim0_stride=512
      const unsigned td0 = (unsigned)W_;       // 512
      const unsigned td1 = 0x7FFFFFFFu;
      u32x8 g1 = {0x00020000u,                              // data_size=2 (4B)
                  (td0 & 0xFFFFu) << 16,                    // abar=0 | td0.lo
                  (td0 >> 16) | ((td1 & 0xFFFFu) << 16),    // td0.hi | td1.lo
                  (td1 >> 16) | ((unsigned)LTW << 16),      // td1.hi | tile0=64
                  (unsigned)LTH,                            // tile1=18 | tile2=0
                  (unsigned)W_,                             // dim0_stride lo32
                  0u, 0u};                                  // stride hi | dim1_stride
      u32x4 gz = {0u, 0u, 0u, 0u};                          // groups 2/3 unused (2D)
      asm volatile("tensor_load_to_lds %0, %1, %2, %3"
                   :: "s"(g0), "s"(g1), "s"(gz), "s"(gz)
                   : "memory");
      tensor_wait0();
    }
  } else {
    // zero-pad border, then per-lane async global->LDS for in-bounds elems
    for (int e = threadIdx.x; e < LTH * LTW; e += blockDim.x) tile[e] = 0.0f;
    __syncthreads();
    for (int e = threadIdx.x; e < LTH * LTW; e += blockDim.x) {
      const int ly = e >> 6, lx = e & 63;
      const int gy = gy0 + ly, gx = gx0 + lx;
      if (gy >= 0 && gy < H_ && gx >= 0 && gx < W_)
        async_g2l_b32(src + gy * W_ + gx, &tile[e]);
    }
    async_wait0();
  }
  __syncthreads();

  const int tid = threadIdx.x;
  const int ox = tid & 63;
  const int oyb = (tid >> 6) << 2;  // 4 rows per thread
  if (ox < CTW) {
    const int x = x0 + ox;
    if (x < W_) {
#pragma unroll
      for (int j = 0; j < 4; ++j) {
        const int oy = oyb + j;
        const float* t0 = &tile[oy * LTW + ox];
        const float a00 = t0[0], a01 = t0[1], a02 = t0[2];
        const float* t1 = t0 + LTW;
        const float a10 = t1[0], a12 = t1[2];
        const float* t2 = t1 + LTW;
        const float a20 = t2[0], a21 = t2[1], a22 = t2[2];
        // XLA conv = cross-correlation (no flip): sx and sx^T
        const float gx = 3.0f * (a02 - a00) + 10.0f * (a12 - a10) + 3.0f * (a22 - a20);
        const float gy = 3.0f * (a20 - a00) + 10.0f * (a21 - a01) + 3.0f * (a22 - a02);
        const int y = y0 + oy;
        out[OFF_GRAD + (plane * 2 + 0) * HW_ + y * W_ + x] = gx;
        out[OFF_GRAD + (plane * 2 + 1) * HW_ + y * W_ + x] = gy;
      }
    }
  }
}

// ---------------- launch ----------------------------------------------------
extern "C" void kernel_launch(void* const* d_in, const int* in_sizes, int n_in,
                              void* d_out, int out_size, void* d_ws, size_t ws_size,
                              hipStream_t stream) {
  (void)in_sizes; (void)n_in; (void)out_size; (void)ws_size;
  const float* img = (const float*)d_in[0];
  const int* imgs_bins = (const int*)d_in[1];
  const int* grads_bins = (const int*)d_in[2];
  const int* reg_lab = (const int*)d_in[3];
  float* out = (float*)d_out;
  int* ws = (int*)d_ws;

  k_init<<<dim3(512), dim3(256), 0, stream>>>(out, ws);
  k_area_bbox<<<dim3(B_ * ACH), dim3(256), 0, stream>>>(reg_lab, ws);
  k_color<<<dim3(B_ * C_ * CCH), dim3(256), (size_t)(S_ * CB_ * 4), stream>>>(
      reg_lab, imgs_bins, out);
  k_texture<<<dim3(B_ * C_ * (R_ / 2) * TCH), dim3(256), (size_t)(2 * S_ * TB_ * 4), stream>>>(
      reg_lab, grads_bins, out);
  k_finalize<<<dim3((B_ * S_ + 255) / 256), dim3(256), 0, stream>>>(ws, out);
  k_normalize<<<dim3(1024), dim3(256), 0, stream>>>(ws, out);
  k_scharr<<<dim3(B_ * C_ * NBX * NBY), dim3(256), 0, stream>>>(img, out);
}